// MultiScaleGraphAttention_24919400251996
// MI455X (gfx1250) — compile-verified
//
#include <hip/hip_runtime.h>
#include <hip/hip_bf16.h>

// Problem constants (from reference)
#define NNODE  50000
#define NC     96
#define NC3    (3 * NC)
#define NEDGE  800000
#define NTILES (NNODE / 16)   // 3125 exactly
#define BN_EPS 1e-5f

typedef float v2f __attribute__((ext_vector_type(2)));
typedef float v8f __attribute__((ext_vector_type(8)));
typedef unsigned int u32x4 __attribute__((ext_vector_type(4)));
typedef int i32x4 __attribute__((ext_vector_type(4)));
typedef int i32x8 __attribute__((ext_vector_type(8)));

// ---------------------------------------------------------------------------
// TDM: 1-D bulk copy global -> LDS via Tensor Data Mover (D# built per ISA
// CDNA5 ISA). count=1, type=2 (image), data_size=4B, one row of
// n elements (n < 65536 fits 16-bit tile_dim0), tensor_dim1/tile_dim1 = 1.
// Issue from ONE wave (EXEC ignored by TDM); caller waits TENSORcnt then
// barriers before any wave reads the LDS.
// ---------------------------------------------------------------------------
__device__ __forceinline__ void tdm_load_1d(const float* gsrc, float* ldst, unsigned n) {
    unsigned long long ga = (unsigned long long)(uintptr_t)gsrc;
    u32x4 g0;
    g0.x = 1u;                                    // count = 1 (valid descriptor)
    g0.y = (unsigned)(uintptr_t)ldst;             // LDS byte address (low 32 bits)
    g0.z = (unsigned)ga;                          // global_addr[31:0]
    g0.w = (unsigned)((ga >> 32) & 0x01ffffffull) // global_addr[56:32]
         | 0x80000000u;                           // type = 2 ("image")
    i32x8 g1;
    g1[0] = (int)(2u << 16);                      // data_size = 2 -> 4 bytes
    g1[1] = (int)((n & 0xffffu) << 16);           // tensor_dim0[15:0]
    g1[2] = (int)((n >> 16) & 0xffffu)            // tensor_dim0[31:16]
          | (1 << 16);                            // tensor_dim1 = 1 (low 16)
    g1[3] = (int)((n & 0xffffu) << 16);           // tile_dim0 = n
    g1[4] = 1;                                    // tile_dim1 = 1, tile_dim2 = 0
    g1[5] = (int)n;                               // tensor_dim0_stride[31:0]
    g1[6] = 0;
    g1[7] = 0;
    i32x4 z4 = {0, 0, 0, 0};
#if defined(__clang_major__) && (__clang_major__ >= 23)
    i32x8 z8 = {0, 0, 0, 0, 0, 0, 0, 0};
    __builtin_amdgcn_tensor_load_to_lds(g0, g1, z4, z4, z8, 0);
#else
    __builtin_amdgcn_tensor_load_to_lds(g0, g1, z4, z4, 0);
#endif
}

// ---------------------------------------------------------------------------
// Utility: zero a region of ints (graph-capture-safe scratch init)
// ---------------------------------------------------------------------------
__global__ void zero_kernel(int* __restrict__ p, int n) {
    int i = blockIdx.x * blockDim.x + threadIdx.x;
    if (i < n) p[i] = 0;
}

// ---------------------------------------------------------------------------
// Degree (over row) for dis[], and per-col counts for CSR build
// ---------------------------------------------------------------------------
__global__ void degree_kernel(const int* __restrict__ row, const int* __restrict__ col,
                              float* __restrict__ deg, int* __restrict__ cnt) {
    int e = blockIdx.x * blockDim.x + threadIdx.x;
    if (e < NEDGE) {
        atomicAdd(&deg[row[e]], 1.0f);
        atomicAdd(&cnt[col[e]], 1);
    }
}

__global__ void dis_kernel(const float* __restrict__ deg, float* __restrict__ dis) {
    int n = blockIdx.x * blockDim.x + threadIdx.x;
    if (n < NNODE) {
        float d = deg[n];
        dis[n] = (d > 0.0f) ? rsqrtf(fmaxf(d, 1.0f)) : 0.0f;
    }
}

// ---------------------------------------------------------------------------
// Single-block exclusive scan of cnt[0..NNODE) -> rowptr[0..NNODE]
// ---------------------------------------------------------------------------
__global__ void scan_kernel(const int* __restrict__ cnt, int* __restrict__ rowptr) {
    __shared__ int sums[1024];
    const int tid = threadIdx.x;
    const int chunk = (NNODE + 1023) / 1024;
    const int begin = tid * chunk;
    const int end = (begin + chunk < NNODE) ? (begin + chunk) : NNODE;

    int s = 0;
    for (int i = begin; i < end; ++i) s += cnt[i];
    sums[tid] = s;
    __syncthreads();

    for (int off = 1; off < 1024; off <<= 1) {
        int v = (tid >= off) ? sums[tid - off] : 0;
        __syncthreads();
        sums[tid] += v;
        __syncthreads();
    }

    int run = (tid == 0) ? 0 : sums[tid - 1];
    for (int i = begin; i < end; ++i) {
        rowptr[i] = run;
        run += cnt[i];
    }
    if (tid == 1023) rowptr[NNODE] = sums[1023];
}

// ---------------------------------------------------------------------------
// Fill CSR (grouped by destination col): src index + edge weight
// ---------------------------------------------------------------------------
__global__ void fill_kernel(const int* __restrict__ row, const int* __restrict__ col,
                            const float* __restrict__ dis,
                            const int* __restrict__ rowptr, int* __restrict__ cursor,
                            int* __restrict__ csr_src, float* __restrict__ csr_w) {
    int e = blockIdx.x * blockDim.x + threadIdx.x;
    if (e < NEDGE) {
        int r = row[e], c = col[e];
        int pos = rowptr[c] + atomicAdd(&cursor[c], 1);
        csr_src[pos] = r;
        csr_w[pos] = -(dis[r] * dis[c]);
    }
}

// ---------------------------------------------------------------------------
// Propagation (gather SpMM): out[n][c] = alpha*sum_e w_e*h[src_e][c] + beta*g[n][c]
// ---------------------------------------------------------------------------
__global__ void prop_kernel(const float* __restrict__ h, const float* __restrict__ g,
                            const int* __restrict__ rowptr,
                            const int* __restrict__ csr_src, const float* __restrict__ csr_w,
                            float alpha, float beta, float* __restrict__ out) {
    int node = blockIdx.x;
    int c = threadIdx.x;
    int s0 = rowptr[node], s1 = rowptr[node + 1];
    float acc = 0.0f;
    for (int e = s0; e < s1; ++e) {
        acc = fmaf(csr_w[e], h[csr_src[e] * NC + c], acc);
    }
    out[node * NC + c] = fmaf(alpha, acc, beta * g[node * NC + c]);
}

// ---------------------------------------------------------------------------
// WMMA fp32 tile compute: one wave -> 16 node rows x 96 cols of
// F = sum_t T_t @ W_t + bias, with W_t staged in LDS.
// k-outer / j-inner: 6 independent accumulator chains (acc[6] = 48 VGPRs),
// A frags loaded once per k-step and reused across all 6 column tiles.
// A (16x4 f32): lane -> row M=lane&15, K = 2*(lane>>4)+{0,1}
// B ( 4x16 f32): lane -> col N=lane&15, K = 2*(lane>>4)+{0,1}
// D (16x16 f32): VGPR v -> M = v + 8*(lane>>4), N = lane&15
// ---------------------------------------------------------------------------
template <int KS>
__device__ __forceinline__ void cheb_compute(const float* const Ts[4],
                                             const float* lw,   // LDS staged weights
                                             const float* __restrict__ bias,
                                             float* __restrict__ F,
                                             int tile, int lane) {
    const int rl = lane & 15;
    const int khalf = lane >> 4;
    const int row = tile * 16 + rl;

    v8f acc[6] = {};
    for (int k0 = 0; k0 < NC; k0 += 4) {
        const int ka = k0 + 2 * khalf;
        v2f a[KS];
        #pragma unroll
        for (int t = 0; t < KS; ++t)
            a[t] = *(const v2f*)(Ts[t] + row * NC + ka);
        #pragma unroll
        for (int j = 0; j < 6; ++j) {
            const int coln = j * 16 + rl;
            #pragma unroll
            for (int t = 0; t < KS; ++t) {
                v2f b;
                b.x = lw[(t * NC + ka) * NC + coln];
                b.y = lw[(t * NC + ka + 1) * NC + coln];
                acc[j] = __builtin_amdgcn_wmma_f32_16x16x4_f32(
                    false, a[t], false, b, (short)0, acc[j], false, false);
            }
        }
    }
    #pragma unroll
    for (int j = 0; j < 6; ++j) {
        const int coln = j * 16 + rl;
        const float bv = bias[coln];
        #pragma unroll
        for (int v = 0; v < 8; ++v)
            F[(tile * 16 + v + 8 * khalf) * NC + coln] = acc[j][v] + bv;
    }
}

// Computes feats for all three scales; weights staged per-scale via TDM.
__global__ void __launch_bounds__(256)
feats_kernel(const float* __restrict__ x,  const float* __restrict__ T1,
             const float* __restrict__ T2, const float* __restrict__ T3,
             const float* __restrict__ w2, const float* __restrict__ b2,
             const float* __restrict__ w3, const float* __restrict__ b3,
             const float* __restrict__ w4, const float* __restrict__ b4,
             float* __restrict__ F2, float* __restrict__ F3, float* __restrict__ F4) {
    __shared__ float ldsw[4 * NC * NC];   // 144 KB: holds one scale's weight stack
    const int lane = threadIdx.x & 31;
    const int wave = threadIdx.x >> 5;
    const int tile = blockIdx.x * (blockDim.x >> 5) + wave;
    const bool live = (tile < NTILES);    // wave-uniform
    const float* Ts[4] = {x, T1, T2, T3};

    // ---- scale K=2 ----
    if (wave == 0) {
        tdm_load_1d(w2, ldsw, 2 * NC * NC);
        __builtin_amdgcn_s_wait_tensorcnt(0);
    }
    __syncthreads();
    if (live) cheb_compute<2>(Ts, ldsw, b2, F2, tile, lane);
    __syncthreads();
    // ---- scale K=3 ----
    if (wave == 0) {
        tdm_load_1d(w3, ldsw, 3 * NC * NC);
        __builtin_amdgcn_s_wait_tensorcnt(0);
    }
    __syncthreads();
    if (live) cheb_compute<3>(Ts, ldsw, b3, F3, tile, lane);
    __syncthreads();
    // ---- scale K=4 ----
    if (wave == 0) {
        tdm_load_1d(w4, ldsw, 4 * NC * NC);
        __builtin_amdgcn_s_wait_tensorcnt(0);
    }
    __syncthreads();
    if (live) cheb_compute<4>(Ts, ldsw, b4, F4, tile, lane);
}

// ---------------------------------------------------------------------------
// Attention: per-node logits over the 288-dim concat, softmax over 3 scales.
// ---------------------------------------------------------------------------
__global__ void attn_kernel(const float* __restrict__ F2, const float* __restrict__ F3,
                            const float* __restrict__ F4,
                            const float* __restrict__ aw, const float* __restrict__ ab,
                            float* __restrict__ SW) {
    const int lane = threadIdx.x & 31;
    const int node = blockIdx.x * (blockDim.x >> 5) + (threadIdx.x >> 5);
    if (node >= NNODE) return;

    float l0 = 0.0f, l1 = 0.0f, l2 = 0.0f;
    for (int c = lane; c < NC3; c += 32) {
        float v;
        if (c < NC)          v = F2[node * NC + c];
        else if (c < 2 * NC) v = F3[node * NC + (c - NC)];
        else                 v = F4[node * NC + (c - 2 * NC)];
        l0 = fmaf(v, aw[c * 3 + 0], l0);
        l1 = fmaf(v, aw[c * 3 + 1], l1);
        l2 = fmaf(v, aw[c * 3 + 2], l2);
    }
    for (int off = 16; off > 0; off >>= 1) {
        l0 += __shfl_xor(l0, off, 32);
        l1 += __shfl_xor(l1, off, 32);
        l2 += __shfl_xor(l2, off, 32);
    }
    l0 += ab[0]; l1 += ab[1]; l2 += ab[2];
    float m = fmaxf(l0, fmaxf(l1, l2));
    float e0 = expf(l0 - m), e1 = expf(l1 - m), e2 = expf(l2 - m);
    float inv = 1.0f / (e0 + e1 + e2);
    if (lane == 0) {
        SW[node * 3 + 0] = e0 * inv;
        SW[node * 3 + 1] = e1 * inv;
        SW[node * 3 + 2] = e2 * inv;
    }
}

// ---------------------------------------------------------------------------
// Fused GEMM: H = [F2*s0 | F3*s1 | F4*s2] @ fus_w + fus_b (WMMA fp32).
// fus_w (288x96, 110.6 KB) staged in LDS via TDM; softmax scale folded into
// the A-fragment load.
// ---------------------------------------------------------------------------
__global__ void __launch_bounds__(256)
fuse_gemm_kernel(const float* __restrict__ F2, const float* __restrict__ F3,
                 const float* __restrict__ F4, const float* __restrict__ SW,
                 const float* __restrict__ fw, const float* __restrict__ fb,
                 float* __restrict__ H) {
    __shared__ float ldsw[NC3 * NC];      // 110.6 KB
    const int lane = threadIdx.x & 31;
    const int wave = threadIdx.x >> 5;
    const int tile = blockIdx.x * (blockDim.x >> 5) + wave;

    if (wave == 0) {
        tdm_load_1d(fw, ldsw, NC3 * NC);
        __builtin_amdgcn_s_wait_tensorcnt(0);
    }
    __syncthreads();
    if (tile >= NTILES) return;           // wave-uniform; no barriers after this

    const int rl = lane & 15;
    const int khalf = lane >> 4;
    const int row = tile * 16 + rl;

    const float* Fs[3] = {F2, F3, F4};
    float sw[3];
    #pragma unroll
    for (int t = 0; t < 3; ++t) sw[t] = SW[row * 3 + t];

    v8f acc[6] = {};
    #pragma unroll
    for (int t = 0; t < 3; ++t) {
        const float* F = Fs[t];
        const float s = sw[t];
        for (int k0 = 0; k0 < NC; k0 += 4) {
            const int ka = k0 + 2 * khalf;
            v2f a = *(const v2f*)(F + row * NC + ka);
            a.x *= s; a.y *= s;
            const int kb = t * NC + ka;
            #pragma unroll
            for (int j = 0; j < 6; ++j) {
                const int coln = j * 16 + rl;
                v2f b;
                b.x = ldsw[kb * NC + coln];
                b.y = ldsw[(kb + 1) * NC + coln];
                acc[j] = __builtin_amdgcn_wmma_f32_16x16x4_f32(
                    false, a, false, b, (short)0, acc[j], false, false);
            }
        }
    }
    #pragma unroll
    for (int j = 0; j < 6; ++j) {
        const int coln = j * 16 + rl;
        const float bv = fb[coln];
        #pragma unroll
        for (int v = 0; v < 8; ++v)
            H[(tile * 16 + v + 8 * khalf) * NC + coln] = acc[j][v] + bv;
    }
}

// ---------------------------------------------------------------------------
// BatchNorm stats + normalize/ReLU
// ---------------------------------------------------------------------------
__global__ void bnsum_kernel(const float* __restrict__ H,
                             float* __restrict__ bsum, float* __restrict__ bsq) {
    const int c = threadIdx.x;           // 96 threads
    float s = 0.0f, s2 = 0.0f;
    for (int n = blockIdx.x; n < NNODE; n += gridDim.x) {
        float v = H[n * NC + c];
        s += v;
        s2 = fmaf(v, v, s2);
    }
    atomicAdd(&bsum[c], s);
    atomicAdd(&bsq[c], s2);
}

__global__ void bnrelu_kernel(const float* __restrict__ H,
                              const float* __restrict__ bsum, const float* __restrict__ bsq,
                              const float* __restrict__ gamma, const float* __restrict__ beta,
                              float* __restrict__ out) {
    int i = blockIdx.x * blockDim.x + threadIdx.x;
    if (i >= NNODE * NC) return;
    int c = i % NC;
    const float invN = 1.0f / (float)NNODE;
    float mean = bsum[c] * invN;
    float var = fmaxf(bsq[c] * invN - mean * mean, 0.0f);
    float y = (H[i] - mean) * rsqrtf(var + BN_EPS) * gamma[c] + beta[c];
    out[i] = fmaxf(y, 0.0f);
}

// ---------------------------------------------------------------------------
// kernel_launch
// ---------------------------------------------------------------------------
extern "C" void kernel_launch(void* const* d_in, const int* in_sizes, int n_in,
                              void* d_out, int out_size, void* d_ws, size_t ws_size,
                              hipStream_t stream) {
    const float* x    = (const float*)d_in[0];
    const int*   ei   = (const int*)  d_in[1];   // [2, E] int32
    const float* w2   = (const float*)d_in[2];
    const float* b2   = (const float*)d_in[3];
    const float* w3   = (const float*)d_in[4];
    const float* b3   = (const float*)d_in[5];
    const float* w4   = (const float*)d_in[6];
    const float* b4   = (const float*)d_in[7];
    const float* aw   = (const float*)d_in[8];
    const float* ab   = (const float*)d_in[9];
    const float* fw   = (const float*)d_in[10];
    const float* fb   = (const float*)d_in[11];
    const float* bng  = (const float*)d_in[12];
    const float* bnb  = (const float*)d_in[13];
    float* out = (float*)d_out;

    const int* row = ei;
    const int* col = ei + NEDGE;

    char* base = (char*)d_ws;
    size_t o = 0;
    auto carve = [&](size_t bytes) -> void* {
        void* p = base + o;
        o = (o + bytes + 255) & ~(size_t)255;
        return p;
    };
    float* deg    = (float*)carve(NNODE * 4);
    int*   cnt    = (int*)  carve((NNODE + 1) * 4);
    int*   cursor = (int*)  carve(NNODE * 4);
    float* bsum   = (float*)carve(NC * 4);
    float* bsq    = (float*)carve(NC * 4);
    const size_t zero_bytes = o;
    float* dis    = (float*)carve(NNODE * 4);
    int*   rowptr = (int*)  carve((NNODE + 1) * 4);
    int*   csrsrc = (int*)  carve((size_t)NEDGE * 4);
    float* csrw   = (float*)carve((size_t)NEDGE * 4);
    float* T1     = (float*)carve((size_t)NNODE * NC * 4);
    float* T2     = (float*)carve((size_t)NNODE * NC * 4);
    float* T3     = (float*)carve((size_t)NNODE * NC * 4);
    float* F2     = (float*)carve((size_t)NNODE * NC * 4);
    float* F3     = (float*)carve((size_t)NNODE * NC * 4);
    float* F4     = (float*)carve((size_t)NNODE * NC * 4);
    float* SW     = (float*)carve((size_t)NNODE * 3 * 4);
    float* H      = T1;                    // T1 dead after feats_kernel -> reuse

    {   // zero counters/accumulators (deterministic under graph replay)
        int nz = (int)(zero_bytes / 4);
        zero_kernel<<<(nz + 255) / 256, 256, 0, stream>>>((int*)d_ws, nz);
    }
    degree_kernel<<<(NEDGE + 255) / 256, 256, 0, stream>>>(row, col, deg, cnt);
    dis_kernel<<<(NNODE + 255) / 256, 256, 0, stream>>>(deg, dis);
    scan_kernel<<<1, 1024, 0, stream>>>(cnt, rowptr);
    fill_kernel<<<(NEDGE + 255) / 256, 256, 0, stream>>>(row, col, dis, rowptr, cursor,
                                                         csrsrc, csrw);
    // Chebyshev basis shared by all scales
    prop_kernel<<<NNODE, NC, 0, stream>>>(x,  x,  rowptr, csrsrc, csrw, 1.0f,  0.0f, T1);
    prop_kernel<<<NNODE, NC, 0, stream>>>(T1, x,  rowptr, csrsrc, csrw, 2.0f, -1.0f, T2);
    prop_kernel<<<NNODE, NC, 0, stream>>>(T2, T1, rowptr, csrsrc, csrw, 2.0f, -1.0f, T3);
    // Nine dense GEMMs on the matrix pipe; weights staged via TDM
    {
        dim3 blk(256), grd((NTILES + 7) / 8);
        feats_kernel<<<grd, blk, 0, stream>>>(x, T1, T2, T3, w2, b2, w3, b3, w4, b4,
                                              F2, F3, F4);
    }
    attn_kernel<<<(NNODE + 7) / 8, 256, 0, stream>>>(F2, F3, F4, aw, ab, SW);
    {
        dim3 blk(256), grd((NTILES + 7) / 8);
        fuse_gemm_kernel<<<grd, blk, 0, stream>>>(F2, F3, F4, SW, fw, fb, H);
    }
    bnsum_kernel<<<512, NC, 0, stream>>>(H, bsum, bsq);
    bnrelu_kernel<<<(NNODE * NC + 255) / 256, 256, 0, stream>>>(H, bsum, bsq, bng, bnb, out);
    (void)in_sizes; (void)n_in; (void)out_size; (void)ws_size;
}